// ResidualBlock_37452114821416
// MI455X (gfx1250) — compile-verified
//
#include <hip/hip_runtime.h>

typedef __attribute__((ext_vector_type(2))) float v2f;
typedef __attribute__((ext_vector_type(8))) float v8f;

#define NV 400000      // active voxels
#define CCH 32         // channels
#define KOFF 27        // 3x3x3 offsets
#define WSTRIDE 34     // padded cin stride (dwords): 2*17, coprime with 32 banks
#define LN_EPS 1e-6f

// One fused pass: out[i,:] = act( LayerNorm( sum_k gather(feats,nbr[k]) @ W[k] ) [+ resid] )
// resid == nullptr  -> ReLU(LN(conv))           (layer 1)
// resid != nullptr  -> ReLU(LN(conv) + resid)   (layer 2 + residual)
__launch_bounds__(256)
__global__ void sconv_ln_wmma(const float* __restrict__ feats,
                              const int*   __restrict__ nbr,
                              const float* __restrict__ W,
                              const float* __restrict__ gamma,
                              const float* __restrict__ beta,
                              const float* __restrict__ resid,
                              float* __restrict__ out)
{
    // Weights staged transposed: sW[k][cout][cin], cin stride padded to 34.
    __shared__ float sW[KOFF * CCH * WSTRIDE];

    const int tid = threadIdx.x;
    // cout fastest -> coalesced global reads; LDS writes stride 34 dwords
    // (lane m -> bank 2*(17*m mod 32): all 32 banks distinct) -> conflict-free.
    for (int i = tid; i < KOFF * CCH * CCH; i += 256) {
        const int cout = i & 31;
        const int cin  = (i >> 5) & 31;
        const int k    = i >> 10;
        sW[(k * CCH + cout) * WSTRIDE + cin] = W[(k * CCH + cin) * CCH + cout];
    }
    __syncthreads();

    const int wave = tid >> 5;
    const int lane = tid & 31;
    const int m    = lane & 15;   // voxel-in-tile (A rows / B cols)
    const int hi   = lane >> 4;   // K-half selector for A/B layouts
    const int base = blockIdx.x * 128 + wave * 16;   // 3125*128 == 400000 exactly
    const int vox  = base + m;

    v8f c0 = {0.f,0.f,0.f,0.f,0.f,0.f,0.f,0.f};   // cout 0..15
    v8f c1 = {0.f,0.f,0.f,0.f,0.f,0.f,0.f,0.f};   // cout 16..31

    for (int k = 0; k < KOFF; ++k) {
        const int   row = nbr[k * NV + vox];
        const float msk = (row >= 0) ? 1.0f : 0.0f;
        const float* rp = feats + (size_t)((row >= 0) ? row : 0) * CCH + 2 * hi;
        const float* wp = sW + (size_t)k * CCH * WSTRIDE + 2 * hi;

#pragma unroll
        for (int kc = 0; kc < 8; ++kc) {
            // A tile 16x4 f32: VGPR0 = K(0|2), VGPR1 = K(1|3) per lane half
            v2f a = *(const v2f*)(rp + kc * 4);
            a *= msk;                                      // zero inactive neighbor, keep EXEC full
            // B tiles 4x16 f32: lane = cout, VGPR0/1 = K(2*hi)/K(2*hi+1)
            v2f b0 = *(const v2f*)(wp + m * WSTRIDE + kc * 4);
            v2f b1 = *(const v2f*)(wp + (16 + m) * WSTRIDE + kc * 4);
            c0 = __builtin_amdgcn_wmma_f32_16x16x4_f32(false, a, false, b0,
                                                       (short)0, c0, false, false);
            c1 = __builtin_amdgcn_wmma_f32_16x16x4_f32(false, a, false, b1,
                                                       (short)0, c1, false, false);
        }
    }

    // Epilogue: LayerNorm across the 32 channels of each voxel.
    // C/D layout: VGPR r, lanes 0-15 -> (M=r,   N=lane),
    //                      lanes 16-31 -> (M=r+8, N=lane-16).
    // Each lane holds one cout for 8 voxels -> reduce across the 16 lanes of a half.
    const float gm0 = gamma[m],      gm1 = gamma[16 + m];
    const float bt0 = beta[m],       bt1 = beta[16 + m];

#pragma unroll
    for (int r = 0; r < 8; ++r) {
        float s = c0[r] + c1[r];
        float q = c0[r] * c0[r] + c1[r] * c1[r];
#pragma unroll
        for (int d = 1; d < 16; d <<= 1) {
            s += __shfl_xor(s, d, 32);
            q += __shfl_xor(q, d, 32);
        }
        const float mu  = s * (1.0f / 32.0f);
        const float var = q * (1.0f / 32.0f) - mu * mu;
        const float rs  = rsqrtf(var + LN_EPS);

        float h0 = (c0[r] - mu) * rs * gm0 + bt0;
        float h1 = (c1[r] - mu) * rs * gm1 + bt1;

        const int    v = base + r + 8 * hi;
        const size_t o = (size_t)v * CCH;
        if (resid != nullptr) {
            h0 += resid[o + m];
            h1 += resid[o + 16 + m];
        }
        h0 = fmaxf(h0, 0.0f);
        h1 = fmaxf(h1, 0.0f);
        out[o + m]      = h0;
        out[o + 16 + m] = h1;
    }
}

extern "C" void kernel_launch(void* const* d_in, const int* in_sizes, int n_in,
                              void* d_out, int out_size, void* d_ws, size_t ws_size,
                              hipStream_t stream) {
    const float* x   = (const float*)d_in[0];
    const int*   nbr = (const int*)  d_in[1];
    const float* W1  = (const float*)d_in[2];
    const float* g1  = (const float*)d_in[3];
    const float* b1  = (const float*)d_in[4];
    const float* W2  = (const float*)d_in[5];
    const float* g2  = (const float*)d_in[6];
    const float* b2  = (const float*)d_in[7];
    float* out = (float*)d_out;
    float* h   = (float*)d_ws;               // intermediate activations: NV*CCH floats

    dim3 grid(NV / 128);                     // 3125 blocks, 8 wave-tiles each
    dim3 block(256);

    // conv1 + LN + ReLU -> h
    sconv_ln_wmma<<<grid, block, 0, stream>>>(x, nbr, W1, g1, b1, nullptr, h);
    // conv2 + LN + residual + ReLU -> out
    sconv_ln_wmma<<<grid, block, 0, stream>>>(h, nbr, W2, g2, b2, x, out);
}